// QuestionSpecificDeepBayesianDKVMN_39857296507064
// MI455X (gfx1250) — compile-verified
//
#include <hip/hip_runtime.h>
#include <hip/hip_bf16.h>
#include <math.h>

// ---- problem constants (from reference) ----
#define NQ   200000
#define KC   4
#define MM   64      // memory slots
#define KEY  64
#define VV   128
#define EE   64
#define BB   512
#define SS   200
#define NROWS (BB*SS)   // 102400

typedef float v2f __attribute__((ext_vector_type(2)));
typedef float v8f __attribute__((ext_vector_type(8)));

static __device__ __forceinline__ v8f wmma_f32(v2f a, v2f b, v8f c) {
    // V_WMMA_F32_16X16X4_F32 : D = A(16x4,f32) x B(4x16,f32) + C(16x16,f32)
    return __builtin_amdgcn_wmma_f32_16x16x4_f32(
        /*neg_a=*/false, a, /*neg_b=*/false, b,
        /*c_mod=*/(short)0, c, /*reuse_a=*/false, /*reuse_b=*/false);
}

static __device__ __forceinline__ float softplusf(float x) {
    return (x > 20.0f) ? x : log1pf(__expf(x));
}

// Async copy 16B global -> LDS (gfx1250 GLOBAL_LOAD_ASYNC_TO_LDS_B128).
// lds_addr: LDS byte address (low 32 bits of flat shared pointer).
static __device__ __forceinline__ void async_g2l_b128(unsigned lds_addr,
                                                      const void* gaddr) {
    asm volatile("global_load_async_to_lds_b128 %0, %1, off"
                 :: "v"(lds_addr), "v"(gaddr) : "memory");
}
static __device__ __forceinline__ void wait_asynccnt0() {
    asm volatile("s_wait_asynccnt 0x0" ::: "memory");
}

// =====================================================================
// Kernel 1: per-row attention via f32 WMMA.
// block = 128 threads (4 waves), each wave owns 16 rows of [B*S].
// =====================================================================
__global__ __launch_bounds__(128)
void k1_attn(const float* __restrict__ q_embed_table,
             const float* __restrict__ memory_keys,
             const float* __restrict__ q_to_key_w,
             const float* __restrict__ q_to_key_b,
             const int*   __restrict__ questions,
             float* __restrict__ attn)              // [NROWS, M]
{
    __shared__ float bufA[4][16][68];   // q_emb tile -> scores tile
    __shared__ float bufB[4][16][68];   // queries tile
    __shared__ int   qix[4][16];

    const int wave = threadIdx.x >> 5;
    const int lane = threadIdx.x & 31;
    const int row  = lane & 15;          // M index within fragment
    const int half = lane >> 4;          // 0: K0/K1 side, 1: K2/K3 side
    const int rowbase = blockIdx.x * 64 + wave * 16;

    if (lane < 16)
        qix[wave][lane] = questions[rowbase + lane];
    __syncthreads();

    // ---- async gather of q_emb tile (16 x 64 f32) straight into LDS ----
    {
        const float* src = q_embed_table + (size_t)qix[wave][row] * EE + half * 32;
        #pragma unroll
        for (int j = 0; j < 8; ++j) {
            unsigned lds = (unsigned)(uintptr_t)&bufA[wave][row][half * 32 + j * 4];
            async_g2l_b128(lds, src + j * 4);
        }
        wait_asynccnt0();
    }
    __syncthreads();

    // ---- queries = q_emb @ q_to_key_w + b : 4 col-tiles x 16 k-steps ----
    #pragma unroll
    for (int ct = 0; ct < 4; ++ct) {
        v8f acc = {};
        const int ncol = ct * 16 + (lane & 15);
        #pragma unroll
        for (int kk = 0; kk < 16; ++kk) {
            const int krow = kk * 4 + half * 2;
            v2f a, b;
            a.x = bufA[wave][row][krow];
            a.y = bufA[wave][row][krow + 1];
            b.x = q_to_key_w[(krow    ) * KEY + ncol];
            b.y = q_to_key_w[(krow + 1) * KEY + ncol];
            acc = wmma_f32(a, b, acc);
        }
        const float bias = q_to_key_b[ncol];
        #pragma unroll
        for (int j = 0; j < 8; ++j)
            bufB[wave][half * 8 + j][ncol] = acc[j] + bias;
    }
    __syncthreads();

    // ---- scores = queries @ memory_keys^T : B[k][m] = memory_keys[m][k] ----
    #pragma unroll
    for (int ct = 0; ct < 4; ++ct) {
        v8f acc = {};
        const int ncol = ct * 16 + (lane & 15);
        #pragma unroll
        for (int kk = 0; kk < 16; ++kk) {
            const int krow = kk * 4 + half * 2;
            v2f a, b;
            a.x = bufB[wave][row][krow];
            a.y = bufB[wave][row][krow + 1];
            b.x = memory_keys[ncol * KEY + krow];
            b.y = memory_keys[ncol * KEY + krow + 1];
            acc = wmma_f32(a, b, acc);
        }
        #pragma unroll
        for (int j = 0; j < 8; ++j)
            bufA[wave][half * 8 + j][ncol] = acc[j];
    }
    __syncthreads();

    // ---- softmax over M=64 (lanes 0..15 each own one row) ----
    if (lane < 16) {
        float mx = -INFINITY;
        #pragma unroll 8
        for (int m = 0; m < MM; ++m) mx = fmaxf(mx, bufA[wave][lane][m]);
        float sum = 0.0f;
        #pragma unroll 8
        for (int m = 0; m < MM; ++m) {
            float e = __expf(bufA[wave][lane][m] - mx);
            bufA[wave][lane][m] = e;
            sum += e;
        }
        const float inv = 1.0f / sum;
        float* dst = attn + (size_t)(rowbase + lane) * MM;
        #pragma unroll 8
        for (int m = 0; m < MM; ++m) dst[m] = bufA[wave][lane][m] * inv;
    }
}

// =====================================================================
// Kernel 2: deterministic batch reductions.
//  wbar[t][m] = (1/B) sum_b attn[b,t,m]
//  hbar[t][e] = (1/B) sum_b tanh(qn*w0[e] + rn*w1[e] + b[e])
// grid = S blocks of 128 threads.
// =====================================================================
__global__ __launch_bounds__(128)
void k2_batch_means(const float* __restrict__ attn,
                    const int*   __restrict__ questions,
                    const int*   __restrict__ responses,
                    const float* __restrict__ qa_embed_w,
                    const float* __restrict__ qa_embed_b,
                    float* __restrict__ wbar,      // [S, M]
                    float* __restrict__ hbar)      // [S, E]
{
    const int t = blockIdx.x;
    const int tid = threadIdx.x;
    if (tid < MM) {
        float s = 0.0f;
        for (int b = 0; b < BB; ++b)
            s += attn[(size_t)(b * SS + t) * MM + tid];
        wbar[t * MM + tid] = s * (1.0f / (float)BB);
    } else {
        const int e = tid - MM;                     // 0..63
        const float w0 = qa_embed_w[e];
        const float w1 = qa_embed_w[EE + e];
        const float bb = qa_embed_b[e];
        float s = 0.0f;
        for (int b = 0; b < BB; ++b) {
            const float qn = (float)questions[b * SS + t] * (1.0f / (float)NQ);
            const float rn = (float)responses[b * SS + t] * (1.0f / (float)(KC - 1));
            s += tanhf(qn * w0 + rn * w1 + bb);
        }
        hbar[t * EE + e] = s * (1.0f / (float)BB);
    }
}

// =====================================================================
// Kernel 2b: evmean[S,V] = hbar[S,E] @ qa_to_ev_w + bias, via f32 WMMA.
// One wave per block; block owns 16 t-rows x 128 cols. 13 blocks.
// =====================================================================
__global__ __launch_bounds__(32)
void k2b_evmean(const float* __restrict__ hbar,
                const float* __restrict__ qa_to_ev_w,
                const float* __restrict__ qa_to_ev_b,
                float* __restrict__ evmean)        // [S, V]
{
    __shared__ float As[16][68];
    const int lane = threadIdx.x;
    const int row  = lane & 15;
    const int half = lane >> 4;
    const int t0   = blockIdx.x * 16;

    // load A tile (clamp duplicate rows past S; they are never stored)
    {
        int t = t0 + row; if (t > SS - 1) t = SS - 1;
        const float4* src = (const float4*)(hbar + (size_t)t * EE + half * 32);
        #pragma unroll
        for (int j = 0; j < 8; ++j) {
            float4 v = src[j];
            int c = half * 32 + j * 4;
            As[row][c+0] = v.x; As[row][c+1] = v.y; As[row][c+2] = v.z; As[row][c+3] = v.w;
        }
    }
    __syncthreads();

    #pragma unroll
    for (int ct = 0; ct < 8; ++ct) {
        v8f acc = {};
        const int ncol = ct * 16 + (lane & 15);
        #pragma unroll
        for (int kk = 0; kk < 16; ++kk) {
            const int krow = kk * 4 + half * 2;
            v2f a, b;
            a.x = As[row][krow];
            a.y = As[row][krow + 1];
            b.x = qa_to_ev_w[(krow    ) * VV + ncol];
            b.y = qa_to_ev_w[(krow + 1) * VV + ncol];
            acc = wmma_f32(a, b, acc);
        }
        const float bias = qa_to_ev_b[ncol];
        #pragma unroll
        for (int j = 0; j < 8; ++j) {
            const int t = t0 + half * 8 + j;
            if (t < SS) evmean[(size_t)t * VV + ncol] = acc[j] + bias;
        }
    }
}

// =====================================================================
// Kernel 3: the (tiny) sequential scan over S. Single block of 256.
// Thread (m = tid>>2, g = tid&3) owns means[m][g*32 .. g*32+31].
// G[t][m] = (1/V) * sum_v means_at_entry(t)[m][v]
// =====================================================================
__global__ __launch_bounds__(256)
void k3_scan(const float* __restrict__ ability_means,
             const float* __restrict__ ability_log_vars,
             const float* __restrict__ wbar,
             const float* __restrict__ evmean,
             float* __restrict__ G)                 // [S, M]
{
    __shared__ float part[256];
    const int tid = threadIdx.x;
    const int m = tid >> 2;
    const int g = tid & 3;

    float mean_r[32], prec_r[32];
    #pragma unroll 8
    for (int j = 0; j < 32; ++j) {
        const int v = g * 32 + j;
        mean_r[j] = ability_means[m * VV + v];
        prec_r[j] = __expf(-ability_log_vars[m * VV + v]);
    }

    for (int t = 0; t < SS; ++t) {
        float s = 0.0f;
        #pragma unroll 8
        for (int j = 0; j < 32; ++j) s += mean_r[j];
        part[tid] = s;
        __syncthreads();
        if (g == 0) {
            G[t * MM + m] = (part[tid] + part[tid + 1] + part[tid + 2] + part[tid + 3])
                            * (1.0f / (float)VV);
        }
        __syncthreads();   // protect part[] before next overwrite

        const float wm = wbar[t * MM + m];
        #pragma unroll 8
        for (int j = 0; j < 32; ++j) {
            const float ev = evmean[t * VV + g * 32 + j];
            const float pp = prec_r[j] + wm;
            mean_r[j] = (prec_r[j] * mean_r[j] + ev * wm) / pp;
            prec_r[j] = pp;
        }
    }
}

// =====================================================================
// Kernel 4: theta = attn_row . G[s]; GPCM softmax over K=4 categories.
// =====================================================================
__global__ __launch_bounds__(256)
void k4_gpcm(const float* __restrict__ attn,
             const float* __restrict__ G,
             const int*   __restrict__ questions,
             const float* __restrict__ alpha_mean,
             const float* __restrict__ beta_base,
             const float* __restrict__ beta_gaps,
             float* __restrict__ out)               // [B, S, K]
{
    const int rrow = blockIdx.x * blockDim.x + threadIdx.x;
    if (rrow >= NROWS) return;
    const int s = rrow % SS;

    const float* ar = attn + (size_t)rrow * MM;
    const float* gr = G + s * MM;
    float th = 0.0f;
    #pragma unroll 8
    for (int m = 0; m < MM; ++m) th += ar[m] * gr[m];

    const int q = questions[rrow];
    const float alpha = __expf(alpha_mean[q]);
    const float b0 = beta_base[q];
    const float g1 = softplusf(beta_gaps[q * (KC - 2) + 0]);
    const float g2 = softplusf(beta_gaps[q * (KC - 2) + 1]);

    float beta[3];
    beta[0] = b0;
    beta[1] = b0 + g1;
    beta[2] = b0 + g1 + g2;

    float cum[4];
    cum[0] = 0.0f;
    #pragma unroll
    for (int k = 1; k < KC; ++k)
        cum[k] = cum[k - 1] + alpha * (th - beta[k - 1]);

    float mx = cum[0];
    #pragma unroll
    for (int k = 1; k < KC; ++k) mx = fmaxf(mx, cum[k]);
    float sum = 0.0f, e[4];
    #pragma unroll
    for (int k = 0; k < KC; ++k) { e[k] = __expf(cum[k] - mx); sum += e[k]; }
    const float inv = 1.0f / sum;
    #pragma unroll
    for (int k = 0; k < KC; ++k) out[(size_t)rrow * KC + k] = e[k] * inv;
}

// =====================================================================
extern "C" void kernel_launch(void* const* d_in, const int* in_sizes, int n_in,
                              void* d_out, int out_size, void* d_ws, size_t ws_size,
                              hipStream_t stream) {
    const float* q_embed_table   = (const float*)d_in[0];
    const float* alpha_mean      = (const float*)d_in[1];
    const float* beta_base       = (const float*)d_in[2];
    const float* beta_gaps       = (const float*)d_in[3];
    const float* ability_means   = (const float*)d_in[4];
    const float* ability_log_vars= (const float*)d_in[5];
    const float* memory_keys     = (const float*)d_in[6];
    const float* q_to_key_w      = (const float*)d_in[7];
    const float* q_to_key_b      = (const float*)d_in[8];
    const float* qa_embed_w      = (const float*)d_in[9];
    const float* qa_embed_b      = (const float*)d_in[10];
    const float* qa_to_ev_w      = (const float*)d_in[11];
    const float* qa_to_ev_b      = (const float*)d_in[12];
    const int*   questions       = (const int*)d_in[13];
    const int*   responses       = (const int*)d_in[14];
    float* out = (float*)d_out;

    float* ws = (float*)d_ws;
    float* attn   = ws;                              // [NROWS*M]
    float* wbar   = attn + (size_t)NROWS * MM;       // [S*M]
    float* hbar   = wbar + SS * MM;                  // [S*E]
    float* evmean = hbar + SS * EE;                  // [S*V]
    float* G      = evmean + SS * VV;                // [S*M]

    // K1: 1600 blocks x 128 threads (each block = 64 rows)
    k1_attn<<<NROWS / 64, 128, 0, stream>>>(
        q_embed_table, memory_keys, q_to_key_w, q_to_key_b, questions, attn);

    // K2: batch means (attn mean + tanh(qa) mean)
    k2_batch_means<<<SS, 128, 0, stream>>>(
        attn, questions, responses, qa_embed_w, qa_embed_b, wbar, hbar);

    // K2b: evmean = hbar @ qa_to_ev_w + b (WMMA, 13 one-wave blocks)
    k2b_evmean<<<(SS + 15) / 16, 32, 0, stream>>>(
        hbar, qa_to_ev_w, qa_to_ev_b, evmean);

    // K3: sequential scan (single block)
    k3_scan<<<1, 256, 0, stream>>>(ability_means, ability_log_vars, wbar, evmean, G);

    // K4: final GPCM probabilities
    k4_gpcm<<<(NROWS + 255) / 256, 256, 0, stream>>>(
        attn, G, questions, alpha_mean, beta_base, beta_gaps, out);
}